// QdDynamics_42425686950244
// MI455X (gfx1250) — compile-verified
//
#include <hip/hip_runtime.h>
#include <math.h>

#ifndef __has_builtin
#define __has_builtin(x) 0
#endif

#define AS1 __attribute__((address_space(1)))
#define AS3 __attribute__((address_space(3)))

typedef int v4i_t __attribute__((vector_size(16)));

// ---------------------------------------------------------------------------
// CDNA5 async global->LDS copy (16B per lane) + asynccnt wait.
// Preferred: clang builtin; fallback: inline asm of the gfx1250 mnemonic.
// ---------------------------------------------------------------------------
__device__ __forceinline__ unsigned __lds_byte_off(void* p) {
  // AS3 pointers are 32-bit LDS offsets on AMDGCN.
  return (unsigned)(uintptr_t)(AS3 void*)p;
}

#if __has_builtin(__builtin_amdgcn_global_load_async_to_lds_b128)
#define ASYNC_COPY_B128(g, l)                                                 \
  __builtin_amdgcn_global_load_async_to_lds_b128((v4i_t*)(g),                 \
                                                 (AS3 v4i_t*)(l), 0, 0)
#else
#warning "builtin global_load_async_to_lds_b128 not found: using inline asm path"
#define ASYNC_COPY_B128(g, l)                                                 \
  asm volatile("global_load_async_to_lds_b128 %0, %1, off"                    \
               :: "v"(__lds_byte_off((void*)(l))),                            \
                  "v"((unsigned long long)(uintptr_t)(g))                     \
               : "memory")
#endif

#if __has_builtin(__builtin_amdgcn_s_wait_asynccnt)
#define ASYNC_WAIT0() __builtin_amdgcn_s_wait_asynccnt(0)
#else
#define ASYNC_WAIT0() asm volatile("s_wait_asynccnt 0" ::: "memory")
#endif

// ---------------------------------------------------------------------------
// Rigid-body derivative, 13-state: [pos(3), v(3), q(4: w,x,y,z), w(3)]
// ---------------------------------------------------------------------------
__device__ __forceinline__ void rb_deriv(const float* x, const float* u, float* k) {
  k[0] = x[3]; k[1] = x[4]; k[2] = x[5];
  k[3] = u[0] / 1.5f;
  k[4] = u[1] / 1.5f;
  k[5] = u[2] / 1.5f + 9.81f;
  const float ew = x[6], ex = x[7], ey = x[8], ez = x[9];
  const float p = x[10], qq = x[11], r = x[12];
  k[6] = 0.5f * (-ex * p - ey * qq - ez * r);
  k[7] = 0.5f * ( ew * p + ey * r  - ez * qq);
  k[8] = 0.5f * ( ew * qq - ex * r + ez * p);
  k[9] = 0.5f * ( ew * r + ex * qq - ey * p);
  const float Jx = 0.0211f, Jy = 0.0219f, Jz = 0.0366f;
  const float jwx = Jx * p, jwy = Jy * qq, jwz = Jz * r;
  const float cx = qq * jwz - r * jwy;
  const float cy = r * jwx - p * jwz;
  const float cz = p * jwy - qq * jwx;
  k[10] = (u[3] - cx) * (1.0f / Jx);
  k[11] = (u[4] - cy) * (1.0f / Jy);
  k[12] = (u[5] - cz) * (1.0f / Jz);
}

__device__ __forceinline__ float clampf(float v, float lo, float hi) {
  return fminf(fmaxf(v, lo), hi);
}

// ---------------------------------------------------------------------------
// One thread per body. 256 bodies per block.
// State staged global->LDS via async b128 (coalesced 16B/lane), padded LDS
// layout (36 floats = 144B per body) so ds_load_b128 readback is conflict-free.
// ---------------------------------------------------------------------------
constexpr int TPB = 256;
constexpr int LSTRIDE = 36;  // floats per body in LDS (padded from 32)

__global__ __launch_bounds__(TPB)
void quad_step_kernel(const float* __restrict__ state,
                      const float* __restrict__ delta,
                      const float* __restrict__ G1,
                      const float* __restrict__ dtp,
                      float* __restrict__ out, int nBodies) {
  __shared__ __align__(16) float smem[TPB * LSTRIDE];  // 36 KB

  const int t = threadIdx.x;
  const long body0 = (long)blockIdx.x * TPB;
  const long totalChunks = (long)nBodies * 8;  // 16B chunks of state

  // ---- Stage: 8 rounds of 256 lanes x 16B, consecutive global addresses ----
#pragma unroll
  for (int rr = 0; rr < 8; ++rr) {
    const int lid = rr * TPB + t;               // local chunk id [0,2048)
    const long gid = body0 * 8 + lid;           // global chunk id
    if (gid < totalChunks) {
      const float* gsrc = state + gid * 4;
      float* ldst = &smem[(lid >> 3) * LSTRIDE + (lid & 7) * 4];
      ASYNC_COPY_B128(gsrc, ldst);
    }
  }
  ASYNC_WAIT0();
  __syncthreads();

  const long body = body0 + t;
  if (body >= nBodies) return;

  // ---- Gather this body's 32 floats from padded LDS (bank-minimal b128) ----
  float s[32];
#pragma unroll
  for (int c = 0; c < 8; ++c) {
    const float4 v = *(const float4*)&smem[t * LSTRIDE + c * 4];
    s[4 * c + 0] = v.x; s[4 * c + 1] = v.y;
    s[4 * c + 2] = v.z; s[4 * c + 3] = v.w;
  }

  const float dt = *dtp;

  // ---- Motor thrust/torque: tt = G1 @ clip(delta)^2 ----
  const float4 dl = *(const float4*)(delta + body * 4);
  const float d0 = clampf(dl.x, 0.f, 1000.f);
  const float d1 = clampf(dl.y, 0.f, 1000.f);
  const float d2 = clampf(dl.z, 0.f, 1000.f);
  const float d3 = clampf(dl.w, 0.f, 1000.f);
  const float q0 = d0 * d0, q1 = d1 * d1, q2 = d2 * d2, q3 = d3 * d3;
  const float T  = G1[0]  * q0 + G1[1]  * q1 + G1[2]  * q2 + G1[3]  * q3;
  const float Mx = G1[4]  * q0 + G1[5]  * q1 + G1[6]  * q2 + G1[7]  * q3;
  const float My = G1[8]  * q0 + G1[9]  * q1 + G1[10] * q2 + G1[11] * q3;
  const float Mz = G1[12] * q0 + G1[13] * q1 + G1[14] * q2 + G1[15] * q3;

  // ---- Rotation matrix from (pre-update) quaternion s[9:13] ----
  const float ew = s[9], ex = s[10], ey = s[11], ez = s[12];
  const float r00 = ew * ew + ex * ex - ey * ey - ez * ez;
  const float r01 = 2.f * (ex * ey - ew * ez);
  const float r02 = 2.f * (ex * ez + ew * ey);
  const float r10 = 2.f * (ex * ey + ew * ez);
  const float r11 = ew * ew - ex * ex + ey * ey - ez * ez;
  const float r12 = 2.f * (ey * ez - ew * ex);
  const float r20 = 2.f * (ex * ez - ew * ey);
  const float r21 = 2.f * (ey * ez + ew * ex);
  const float r22 = ew * ew - ex * ex - ey * ey + ez * ez;

  // ---- Relative wind (body frame): V_wind_b = R^T * wind_i ----
  const float wxi = s[28], wyi = s[29], wzi = s[30];
  const float vb0 = r00 * wxi + r10 * wyi + r20 * wzi;
  const float vb1 = r01 * wxi + r11 * wyi + r21 * wzi;
  const float vb2 = r02 * wxi + r12 * wyi + r22 * wzi;
  const float u_r = s[16] - vb0;
  const float v_r = s[17] - vb1;
  const float w_r = s[18] - vb2;
  const float Va = sqrtf(u_r * u_r + v_r * v_r + w_r * w_r);
  const float alpha = (u_r == 0.f) ? 1.57079632679489662f : atan2f(w_r, u_r);
  const float beta  = (Va == 0.f) ? 0.f : asinf(clampf(v_r / Va, -1.f, 1.f));

  // ---- Forces (body), rotate to inertial ----
  const float fx = -0.1f * u_r;
  const float fy = -0.1f * v_r;
  const float fz = -T - 0.1f * w_r + 0.01f * (u_r * u_r + v_r * v_r);
  const float fi0 = r00 * fx + r01 * fy + r02 * fz;
  const float fi1 = r10 * fx + r11 * fy + r12 * fz;
  const float fi2 = r20 * fx + r21 * fy + r22 * fz;

  // ---- RK4 on the 13-state ODE ----
  const float uu[6] = {fi0, fi1, fi2, Mx, My, Mz};
  float x0[13] = {s[3],  s[4],  s[5],
                  s[13], s[14], s[15],
                  s[9],  s[10], s[11], s[12],
                  s[19], s[20], s[21]};
  float kk[13], acc[13], xt[13], xn[13];
  const float hdt = 0.5f * dt;

  rb_deriv(x0, uu, kk);
#pragma unroll
  for (int i = 0; i < 13; ++i) { acc[i] = kk[i]; xt[i] = x0[i] + hdt * kk[i]; }
  rb_deriv(xt, uu, kk);
#pragma unroll
  for (int i = 0; i < 13; ++i) { acc[i] += 2.f * kk[i]; xt[i] = x0[i] + hdt * kk[i]; }
  rb_deriv(xt, uu, kk);
#pragma unroll
  for (int i = 0; i < 13; ++i) { acc[i] += 2.f * kk[i]; xt[i] = x0[i] + dt * kk[i]; }
  rb_deriv(xt, uu, kk);
  const float dt6 = dt / 6.0f;
#pragma unroll
  for (int i = 0; i < 13; ++i) xn[i] = x0[i] + dt6 * (acc[i] + kk[i]);

  // ---- Normalize updated quaternion ----
  const float qn2 = xn[6] * xn[6] + xn[7] * xn[7] + xn[8] * xn[8] + xn[9] * xn[9];
  const float qnrm = sqrtf(qn2);
  const float nw = xn[6] / qnrm, nx = xn[7] / qnrm;
  const float ny = xn[8] / qnrm, nz = xn[9] / qnrm;

  // ---- Euler from the NEW quaternion ----
  const float phi   = atan2f(2.f * (nw * nx + ny * nz),
                             nw * nw + nz * nz - nx * nx - ny * ny);
  const float theta = asinf(clampf(2.f * (nw * ny - nx * nz), -1.f, 1.f));
  const float psi   = atan2f(2.f * (nw * nz + nx * ny),
                             nw * nw + nx * nx - ny * ny - nz * nz);

  // ---- Ground speed / path angles from OLD R and body velocity s[16:19] ----
  const float p0 = r00 * s[16] + r01 * s[17] + r02 * s[18];
  const float p1 = r10 * s[16] + r11 * s[17] + r12 * s[18];
  const float p2 = r20 * s[16] + r21 * s[17] + r22 * s[18];
  const float Vg = sqrtf(p0 * p0 + p1 * p1 + p2 * p2);
  const float Vgs = (Vg == 0.f) ? 1.f : Vg;
  const float gamma = asinf(clampf(p2 / Vgs, -1.f, 1.f));
  const float chi = atan2f(p1, p0);

  // ---- Assemble output state ----
  float o[32];
  o[0] = s[0];  o[1] = s[1];  o[2] = s[2];
  o[3] = xn[0]; o[4] = xn[1]; o[5] = xn[2];
  o[6] = phi;   o[7] = theta; o[8] = psi;
  o[9] = nw;    o[10] = nx;   o[11] = ny;  o[12] = nz;
  o[13] = xn[3]; o[14] = xn[4]; o[15] = xn[5];
  o[16] = s[16]; o[17] = s[17]; o[18] = s[18];
  o[19] = xn[10]; o[20] = xn[11]; o[21] = xn[12];
  o[22] = Va;   o[23] = Vg;  o[24] = alpha; o[25] = beta;
  o[26] = gamma; o[27] = chi;
  o[28] = s[28]; o[29] = s[29]; o[30] = s[30]; o[31] = s[31];

  float4* ov = (float4*)(out + body * 32);
#pragma unroll
  for (int c = 0; c < 8; ++c) {
    ov[c] = make_float4(o[4 * c + 0], o[4 * c + 1], o[4 * c + 2], o[4 * c + 3]);
  }
}

extern "C" void kernel_launch(void* const* d_in, const int* in_sizes, int n_in,
                              void* d_out, int out_size, void* d_ws, size_t ws_size,
                              hipStream_t stream) {
  const float* state = (const float*)d_in[0];
  const float* delta = (const float*)d_in[1];
  const float* G1    = (const float*)d_in[2];
  const float* dtp   = (const float*)d_in[3];
  float* out = (float*)d_out;
  const int nBodies = in_sizes[0] / 32;
  const int blocks = (nBodies + TPB - 1) / TPB;
  quad_step_kernel<<<blocks, TPB, 0, stream>>>(state, delta, G1, dtp, out, nBodies);
  (void)n_in; (void)out_size; (void)d_ws; (void)ws_size;
}